// PolyMPNN_919123001905
// MI455X (gfx1250) — compile-verified
//
#include <hip/hip_runtime.h>

// ---------------------------------------------------------------------------
// PolyMPNN for MI455X (gfx1250): fp32 end-to-end using V_WMMA_F32_16X16X4_F32.
// Orientation: D[ch x edges] = W^T[ch x K] * msg^T[K x edges] so edge gathers
// feed the B operand directly (8-byte pair loads matching the 32-bit WMMA
// operand layout). h (12.8MB) is L2-resident; kernel is L2-gather/atomic bound.
// ---------------------------------------------------------------------------

typedef __attribute__((ext_vector_type(2))) float v2f;
typedef __attribute__((ext_vector_type(8))) float v8f;

#define EMBED   64
#define HIDDEN  128
#define NLAYERS 4
#define POLY    8
#define KIN     130      // 2*EMBED + EDGE_FEAT
#define KPAD    132      // padded to multiple of 4 for K=4 WMMA steps

__device__ __forceinline__ v8f wmma_f32(v2f a, v2f b, v8f c) {
  // (neg_a, A, neg_b, B, c_mod, C, reuse_a, reuse_b)
  return __builtin_amdgcn_wmma_f32_16x16x4_f32(false, a, false, b, (short)0, c,
                                               false, false);
}

__device__ __forceinline__ float wave_sum(float v) {
#pragma unroll
  for (int m = 16; m >= 1; m >>= 1) v += __shfl_xor(v, m, 32);
  return v;
}

// ---------------------------------------------------------------------------
// Prep: transpose weights once so WMMA A-fragments are contiguous 8B pairs.
//   w1T[l][ch][k] (k padded to 132, zeros in pad)   <- conv_w1[l][k][ch]
//   w2T[l][ch][k]                                   <- conv_w2[l][k][ch]
//   swT[l][ch][k]                                   <- skip_w[l][k][ch]
// ---------------------------------------------------------------------------
__global__ void prep_kernel(const float* __restrict__ cw1,
                            const float* __restrict__ cw2,
                            const float* __restrict__ sw,
                            float* __restrict__ w1T, float* __restrict__ w2T,
                            float* __restrict__ swT) {
  int idx = blockIdx.x * blockDim.x + threadIdx.x;
  const int S1 = NLAYERS * EMBED * KPAD;   // 33792
  const int S2 = NLAYERS * EMBED * EMBED;  // 16384
  if (idx < S1) {
    int l = idx / (EMBED * KPAD), rem = idx % (EMBED * KPAD);
    int ch = rem / KPAD, k = rem % KPAD;
    w1T[idx] = (k < KIN) ? cw1[(l * KIN + k) * EMBED + ch] : 0.f;
  } else if (idx < S1 + S2) {
    int i = idx - S1;
    int l = i / (EMBED * EMBED), rem = i % (EMBED * EMBED);
    int ch = rem / EMBED, k = rem % EMBED;
    w2T[i] = cw2[(l * EMBED + k) * EMBED + ch];
  } else if (idx < S1 + 2 * S2) {
    int i = idx - S1 - S2;
    int l = i / (EMBED * EMBED), rem = i % (EMBED * EMBED);
    int ch = rem / EMBED, k = rem % EMBED;
    swT[i] = sw[(l * EMBED + k) * EMBED + ch];
  }
}

// ---------------------------------------------------------------------------
// Node encoder: h = relu(x @ W1[3x128] + b1) @ W2[128x64] + b2.
// One wave32 per node; hidden staged in per-wave LDS (intra-wave DS is
// in-order; s_wait_dscnt 0 orders the cross-lane RAW).
// ---------------------------------------------------------------------------
__global__ void enc_kernel(const float* __restrict__ x,
                           const float* __restrict__ w1,
                           const float* __restrict__ b1,
                           const float* __restrict__ w2,
                           const float* __restrict__ b2,
                           float* __restrict__ h, int n) {
  __shared__ float hidS[8][HIDDEN];
  int wave = threadIdx.x >> 5, lane = threadIdx.x & 31;
  int node = blockIdx.x * 8 + wave;
  if (node >= n) return;
  float x0 = x[node * 3 + 0], x1 = x[node * 3 + 1], x2 = x[node * 3 + 2];
#pragma unroll
  for (int i = 0; i < 4; ++i) {
    int c = lane + 32 * i;
    float v = fmaf(x0, w1[c],
              fmaf(x1, w1[HIDDEN + c],
              fmaf(x2, w1[2 * HIDDEN + c], b1[c])));
    hidS[wave][c] = v > 0.f ? v : 0.f;
  }
  asm volatile("s_wait_dscnt 0" ::: "memory");
#pragma unroll
  for (int i = 0; i < 2; ++i) {
    int c = lane + 32 * i;
    float acc = b2[c];
    for (int k = 0; k < HIDDEN; ++k) acc = fmaf(hidS[wave][k], w2[k * EMBED + c], acc);
    h[(size_t)node * EMBED + c] = acc;
  }
}

// ---------------------------------------------------------------------------
// Skip GEMM via WMMA: agg[node][ch] = (h @ skip_w + skip_b).
// D[ch x nodes] = swT[ch x K] * h^T[K x nodes], one wave per 16-node tile.
// Pre-fills the segment_sum target so no memset pass is needed.
// ---------------------------------------------------------------------------
__global__ void skip_kernel(const float* __restrict__ h,
                            const float* __restrict__ swT,
                            const float* __restrict__ sb,
                            float* __restrict__ agg, int n) {
  int lane = threadIdx.x & 31, wave = threadIdx.x >> 5;
  int tile = blockIdx.x * 8 + wave;
  if (tile * 16 >= n) return;                 // wave-uniform: EXEC stays full
  int hi = lane >> 4, l16 = lane & 15;
  int node = tile * 16 + l16;
  bool valid = node < n;
  int nodec = valid ? node : (n - 1);         // clamp for safe loads
  const float* hrow = h + (size_t)nodec * EMBED;

  v8f acc[4];
#pragma unroll
  for (int m = 0; m < 4; ++m)
#pragma unroll
    for (int j = 0; j < 8; ++j) acc[m][j] = sb[m * 16 + j + hi * 8];

#pragma unroll
  for (int t = 0; t < 16; ++t) {
    int k = 4 * t + 2 * hi;
    v2f b = *(const v2f*)(hrow + k);
#pragma unroll
    for (int m = 0; m < 4; ++m) {
      v2f a = *(const v2f*)(swT + (m * 16 + l16) * EMBED + k);
      acc[m] = wmma_f32(a, b, acc[m]);
    }
  }
  if (valid) {
#pragma unroll
    for (int m = 0; m < 4; ++m)
#pragma unroll
      for (int j = 0; j < 8; ++j)
        agg[(size_t)node * EMBED + m * 16 + j + hi * 8] = acc[m][j];
  }
}

// ---------------------------------------------------------------------------
// Edge message MLP + scatter. One wave per 16-edge tile:
//   mid = relu(W1^T[64x132] * msg^T[132x16] + b1)   (33 K-steps x 4 ch-tiles)
//   out = W2^T[64x64] * mid[64x16] + b2             (16 K-steps x 4 ch-tiles)
//   atomicAdd into agg[row].
// msg^T columns gathered straight from h[rows]/h[cols]/edge_features.
// mid transposed edge-major through per-wave LDS (stride 68 to spread banks).
// ---------------------------------------------------------------------------
__global__ void edge_kernel(const float* __restrict__ h,
                            const int* __restrict__ rows,
                            const int* __restrict__ cols,
                            const float* __restrict__ ef,
                            const float* __restrict__ w1T,
                            const float* __restrict__ w2T,
                            const float* __restrict__ b1,
                            const float* __restrict__ b2,
                            float* __restrict__ agg, int n_edges) {
  __shared__ float midS[8][16 * 68];
  int lane = threadIdx.x & 31, wave = threadIdx.x >> 5;
  int tile = blockIdx.x * 8 + wave;
  if (tile * 16 >= n_edges) return;           // wave-uniform: EXEC stays full
  int hi = lane >> 4, l16 = lane & 15;
  int e = tile * 16 + l16;
  bool valid = e < n_edges;
  int ec = valid ? e : (n_edges - 1);
  int r = rows[ec], c = cols[ec];
  const float* hr = h + (size_t)r * EMBED;
  const float* hc = h + (size_t)c * EMBED;

  v8f acc[4];
#pragma unroll
  for (int m = 0; m < 4; ++m)
#pragma unroll
    for (int j = 0; j < 8; ++j) acc[m][j] = b1[m * 16 + j + hi * 8];

  // --- K segment 1: features 0..63 = h[row] ---
#pragma unroll
  for (int t = 0; t < 16; ++t) {
    int k = 4 * t + 2 * hi;
    v2f b = *(const v2f*)(hr + k);
#pragma unroll
    for (int m = 0; m < 4; ++m) {
      v2f a = *(const v2f*)(w1T + (m * 16 + l16) * KPAD + k);
      acc[m] = wmma_f32(a, b, acc[m]);
    }
  }
  // --- K segment 2: features 64..127 = h[col] ---
#pragma unroll
  for (int t = 16; t < 32; ++t) {
    int k = 4 * t + 2 * hi;
    v2f b = *(const v2f*)(hc + (k - EMBED));
#pragma unroll
    for (int m = 0; m < 4; ++m) {
      v2f a = *(const v2f*)(w1T + (m * 16 + l16) * KPAD + k);
      acc[m] = wmma_f32(a, b, acc[m]);
    }
  }
  // --- K segment 3: features 128..129 = edge_features, 130..131 = zero pad ---
  {
    int k = 128 + 2 * hi;
    v2f b;
    if (hi) { b.x = 0.f; b.y = 0.f; }
    else    { b.x = ef[2 * (size_t)ec]; b.y = ef[2 * (size_t)ec + 1]; }
#pragma unroll
    for (int m = 0; m < 4; ++m) {
      v2f a = *(const v2f*)(w1T + (m * 16 + l16) * KPAD + k);  // pad cols are 0
      acc[m] = wmma_f32(a, b, acc[m]);
    }
  }

  // relu + transpose mid (ch-major -> edge-major) through per-wave LDS
  float* my = midS[wave];
#pragma unroll
  for (int m = 0; m < 4; ++m)
#pragma unroll
    for (int j = 0; j < 8; ++j) {
      int ch = m * 16 + j + hi * 8;
      float v = acc[m][j];
      my[l16 * 68 + ch] = v > 0.f ? v : 0.f;
    }
  asm volatile("s_wait_dscnt 0" ::: "memory");  // intra-wave DS RAW ordering

  v8f o[4];
#pragma unroll
  for (int m = 0; m < 4; ++m)
#pragma unroll
    for (int j = 0; j < 8; ++j) o[m][j] = b2[m * 16 + j + hi * 8];

#pragma unroll
  for (int t = 0; t < 16; ++t) {
    int k = 4 * t + 2 * hi;
    v2f b = *(const v2f*)(my + l16 * 68 + k);
#pragma unroll
    for (int m = 0; m < 4; ++m) {
      v2f a = *(const v2f*)(w2T + (m * 16 + l16) * EMBED + k);
      o[m] = wmma_f32(a, b, o[m]);
    }
  }

  // scatter-add (segment_sum over source rows): lane owns edge l16's channels
  if (valid) {
#pragma unroll
    for (int m = 0; m < 4; ++m)
#pragma unroll
      for (int j = 0; j < 8; ++j)
        atomicAdd(agg + (size_t)r * EMBED + m * 16 + j + hi * 8, o[m][j]);
  }
}

// ---------------------------------------------------------------------------
// LayerNorm + ReLU, in place. One wave per node, 2 channels/lane, shfl reduce.
// ---------------------------------------------------------------------------
__global__ void ln_kernel(float* __restrict__ hio, const float* __restrict__ g,
                          const float* __restrict__ b, int n) {
  int wave = threadIdx.x >> 5, lane = threadIdx.x & 31;
  int node = blockIdx.x * 8 + wave;
  if (node >= n) return;
  float* row = hio + (size_t)node * EMBED;
  float x0 = row[lane], x1 = row[lane + 32];
  float s  = wave_sum(x0 + x1);
  float ss = wave_sum(x0 * x0 + x1 * x1);
  float mu = s * (1.f / 64.f);
  float var = ss * (1.f / 64.f) - mu * mu;
  float rs = rsqrtf(var + 1e-5f);
  float y0 = (x0 - mu) * rs * g[lane] + b[lane];
  float y1 = (x1 - mu) * rs * g[lane + 32] + b[lane + 32];
  row[lane]      = y0 > 0.f ? y0 : 0.f;
  row[lane + 32] = y1 > 0.f ? y1 : 0.f;
}

// ---------------------------------------------------------------------------
// Polynomial head: out = relu(h @ W1[64x64] + b1) @ W2[64x8] + b2.
// One wave per node: 2 hidden ch/lane, shfl-reduce the 8 outputs.
// ---------------------------------------------------------------------------
__global__ void head_kernel(const float* __restrict__ h,
                            const float* __restrict__ w1,
                            const float* __restrict__ b1,
                            const float* __restrict__ w2,
                            const float* __restrict__ b2,
                            float* __restrict__ out, int n) {
  int wave = threadIdx.x >> 5, lane = threadIdx.x & 31;
  int node = blockIdx.x * 8 + wave;
  if (node >= n) return;
  const float* row = h + (size_t)node * EMBED;
  float a0 = b1[lane], a1 = b1[lane + 32];
  for (int k = 0; k < EMBED; ++k) {
    float hv = row[k];
    a0 = fmaf(hv, w1[k * EMBED + lane], a0);
    a1 = fmaf(hv, w1[k * EMBED + lane + 32], a1);
  }
  a0 = a0 > 0.f ? a0 : 0.f;
  a1 = a1 > 0.f ? a1 : 0.f;
  float p[POLY];
#pragma unroll
  for (int j = 0; j < POLY; ++j)
    p[j] = a0 * w2[lane * POLY + j] + a1 * w2[(lane + 32) * POLY + j];
#pragma unroll
  for (int j = 0; j < POLY; ++j) p[j] = wave_sum(p[j]);
  if (lane == 0) {
#pragma unroll
    for (int j = 0; j < POLY; ++j) out[(size_t)node * POLY + j] = p[j] + b2[j];
  }
}

// ---------------------------------------------------------------------------
extern "C" void kernel_launch(void* const* d_in, const int* in_sizes, int n_in,
                              void* d_out, int out_size, void* d_ws, size_t ws_size,
                              hipStream_t stream) {
  const float* x      = (const float*)d_in[0];
  const int*   ei     = (const int*)  d_in[1];
  const float* ef     = (const float*)d_in[2];
  const float* enc_w1 = (const float*)d_in[4];
  const float* enc_b1 = (const float*)d_in[5];
  const float* enc_w2 = (const float*)d_in[6];
  const float* enc_b2 = (const float*)d_in[7];
  const float* cw1    = (const float*)d_in[8];
  const float* cb1    = (const float*)d_in[9];
  const float* cw2    = (const float*)d_in[10];
  const float* cb2    = (const float*)d_in[11];
  const float* sw     = (const float*)d_in[12];
  const float* sb     = (const float*)d_in[13];
  const float* lg     = (const float*)d_in[14];
  const float* lb     = (const float*)d_in[15];
  const float* hw1    = (const float*)d_in[16];
  const float* hb1    = (const float*)d_in[17];
  const float* hw2    = (const float*)d_in[18];
  const float* hb2    = (const float*)d_in[19];

  const int n  = in_sizes[0] / 3;   // nodes
  const int ne = in_sizes[2] / 2;   // edges
  const int* rows = ei;
  const int* cols = ei + ne;

  // workspace layout (floats): two ping-pong h buffers + transposed weights
  float* ws = (float*)d_ws;
  size_t hsz = (size_t)n * EMBED;
  float* hA  = ws;
  float* hB  = ws + hsz;
  float* w1T = ws + 2 * hsz;
  float* w2T = w1T + NLAYERS * EMBED * KPAD;
  float* swT = w2T + NLAYERS * EMBED * EMBED;

  const int prepN = NLAYERS * EMBED * KPAD + 2 * NLAYERS * EMBED * EMBED;
  prep_kernel<<<(prepN + 255) / 256, 256, 0, stream>>>(cw1, cw2, sw, w1T, w2T, swT);

  const int nodeBlocks = (n + 7) / 8;                       // 8 waves/block
  enc_kernel<<<nodeBlocks, 256, 0, stream>>>(x, enc_w1, enc_b1, enc_w2, enc_b2, hA, n);

  const int ntiles = (n + 15) / 16;
  const int skipBlocks = (ntiles + 7) / 8;
  const int etiles = (ne + 15) / 16;
  const int edgeBlocks = (etiles + 7) / 8;

  float* hc = hA;
  float* ho = hB;
  for (int l = 0; l < NLAYERS; ++l) {
    skip_kernel<<<skipBlocks, 256, 0, stream>>>(hc, swT + l * EMBED * EMBED,
                                                sb + l * EMBED, ho, n);
    edge_kernel<<<edgeBlocks, 256, 0, stream>>>(hc, rows, cols, ef,
                                                w1T + l * EMBED * KPAD,
                                                w2T + l * EMBED * EMBED,
                                                cb1 + l * EMBED, cb2 + l * EMBED,
                                                ho, ne);
    ln_kernel<<<nodeBlocks, 256, 0, stream>>>(ho, lg + l * EMBED, lb + l * EMBED, n);
    float* t = hc; hc = ho; ho = t;
  }

  head_kernel<<<nodeBlocks, 256, 0, stream>>>(hc, hw1, hb1, hw2, hb2,
                                              (float*)d_out, n);
}